// MyModel_87522843560086
// MI455X (gfx1250) — compile-verified
//
#include <hip/hip_runtime.h>
#include <stdint.h>

// ---------------------------------------------------------------------------
// Greedy NMS (tf.image.non_max_suppression semantics) as a single persistent
// kernel on gfx1250 (MI455X).
//
// MI455X reasoning:
//  * 20 MB working set (live scores + canonical boxes) << 192 MB L2 -> all
//    300 sequential rounds run out of L2; HBM is touched once (prologue).
//  * The 300 rounds are latency-bound, so: persistent kernel + exactly ONE
//    software grid sync per round. Each block publishes its best candidate
//    with a monotone u64 atomicMax key (score bits || ~index, so equal
//    scores resolve to the lowest index, matching jnp.argmax).
//  * CDNA5 data path: double-buffered GLOBAL_LOAD_ASYNC_TO_LDS_B128/_B32
//    (ASYNCcnt + s_wait_asynccnt) streams box/score tiles into LDS; wave32
//    shuffle butterflies reduce; device-scope atomics publish; s_sleep spin.
//  * Tiles are block- and lane-private, so inter-round coherence needs only
//    a per-wave s_wait_storecnt drain before each barrier — no DEV-scope
//    writeback/invalidate on the critical path.
// ---------------------------------------------------------------------------

#define NB    128          // persistent blocks (co-resident: 128*8 = 1024 wave32)
#define BS    256          // threads per block (8 waves)
#define NW    (BS / 32)
#define MAXB  300
#define SCORE_THR 0.1f
#define NEGINF (-__builtin_huge_valf())

typedef unsigned long long u64;

// Monotone (score, lowest-index-wins) key for u64 atomicMax argmax.
__device__ __forceinline__ u64 enc_key(float v, int i) {
  unsigned u = __float_as_uint(v);
  u = (u & 0x80000000u) ? ~u : (u | 0x80000000u);
  return ((u64)u << 32) | (u64)(~(unsigned)i);
}
__device__ __forceinline__ void dec_key(u64 k, float& v, int& i) {
  unsigned u = (unsigned)(k >> 32);
  u = (u & 0x80000000u) ? (u ^ 0x80000000u) : ~u;
  v = __uint_as_float(u);
  i = (int)(~(unsigned)k);
}

// Low 32 bits of a flat pointer into LDS == offset within the wave's LDS
// allocation (flat-aperture LDS mapping, ISA 10.2).
template <typename T>
__device__ __forceinline__ unsigned lds_off(T* p) {
  return (unsigned)(uintptr_t)(void*)p;
}

// Software grid barrier (graph-capture safe; no host APIs).
// Per-wave storecnt drain is all the publication this algorithm needs:
// every global store is re-read only by the wave that issued it (tile data)
// or is an agent-scope atomic (keys/bar) ordered by the barrier RMWs.
__device__ __forceinline__ void gsync(unsigned* cnt, unsigned* sen) {
  asm volatile("s_wait_storecnt 0x0" ::: "memory");
  __syncthreads();
  if (threadIdx.x == 0) {
    unsigned g = __hip_atomic_load(sen, __ATOMIC_RELAXED, __HIP_MEMORY_SCOPE_AGENT);
    if (__hip_atomic_fetch_add(cnt, 1u, __ATOMIC_ACQ_REL, __HIP_MEMORY_SCOPE_AGENT) == NB - 1u) {
      __hip_atomic_store(cnt, 0u, __ATOMIC_RELAXED, __HIP_MEMORY_SCOPE_AGENT);
      __hip_atomic_fetch_add(sen, 1u, __ATOMIC_RELEASE, __HIP_MEMORY_SCOPE_AGENT);
    } else {
      while (__hip_atomic_load(sen, __ATOMIC_ACQUIRE, __HIP_MEMORY_SCOPE_AGENT) == g) {
        __builtin_amdgcn_s_sleep(2);
      }
    }
  }
  __syncthreads();
}

// Issue CDNA5 async global->LDS copies for one 256-element tile
// (float4 canonical box + float score per lane). Tracked with ASYNCcnt.
__device__ __forceinline__ void stage_tile(u64 cb_base, u64 s_base, int tile,
                                           unsigned ldsCb, unsigned ldsS) {
  unsigned i = (unsigned)tile * BS + threadIdx.x;
  unsigned offCb = i * 16u;
  unsigned offS  = i * 4u;
  asm volatile("global_load_async_to_lds_b128 %0, %1, %2"
               :: "v"(ldsCb), "v"(offCb), "s"(cb_base) : "memory");
  asm volatile("global_load_async_to_lds_b32 %0, %1, %2"
               :: "v"(ldsS), "v"(offS), "s"(s_base) : "memory");
}

__global__ void __launch_bounds__(BS) nms_init_kernel(u64* keys, unsigned* bar) {
  int i = threadIdx.x;
  if (i < MAXB) keys[i] = 0ull;
  if (i == 0) { bar[0] = 0u; bar[1] = 0u; }
}

__global__ void __launch_bounds__(BS) nms_persistent_kernel(
    const float* __restrict__ scores, const float4* __restrict__ boxes,
    const int* __restrict__ classes, float* __restrict__ out,
    float* __restrict__ s, float4* __restrict__ cb,
    u64* __restrict__ keys, unsigned* __restrict__ bar,
    int n, int tilesTotal) {
  __shared__ float4 cbT[2][BS];          // 8 KB double buffer, async-filled
  __shared__ float  sT[2][BS];           // 2 KB double buffer, async-filled
  __shared__ u64    red[NW];
  __shared__ float  sel[5];              // y1,x1,y2,x2,area of selected box
  __shared__ int    selOk;

  const int tid  = threadIdx.x;
  const int npad = tilesTotal * BS;
  const u64 cb_base = (u64)(uintptr_t)cb;
  const u64 s_base  = (u64)(uintptr_t)s;

  // ---- Prologue: thresholded scores + canonicalized boxes, -inf padding.
  // Stride NB*BS => each block initializes exactly its own tiles
  // (tile ≡ blockIdx mod NB): all tile data stays block/lane-private.
  for (int i = blockIdx.x * BS + tid; i < npad; i += NB * BS) {
    if (i < n) {
      float4 b = boxes[i];                       // raw [y1,x1,y2,x2]
      float y1 = fminf(b.x, b.z), y2 = fmaxf(b.x, b.z);
      float x1 = fminf(b.y, b.w), x2 = fmaxf(b.y, b.w);
      cb[i] = make_float4(y1, x1, y2, x2);
      float sc = scores[i];
      s[i] = (sc > SCORE_THR) ? sc : NEGINF;
    } else {
      cb[i] = make_float4(0.f, 0.f, 0.f, 0.f);
      s[i]  = NEGINF;
    }
  }
  gsync(bar, bar + 1);

  bool  prevOk = false;
  float by1 = 0.f, bx1 = 0.f, by2 = 0.f, bx2 = 0.f, ba = 0.f;

  for (int t = 0; t < MAXB; ++t) {
    // ---- Fused (suppress-by-previous-pick + argmax) sweep over own tiles ----
    u64 best = 0ull;
    int tile = blockIdx.x;
    int buf  = 0;
    if (tile < tilesTotal)
      stage_tile(cb_base, s_base, tile, lds_off(&cbT[0][tid]), lds_off(&sT[0][tid]));
    while (tile < tilesTotal) {
      int nt = tile + NB;
      if (nt < tilesTotal) {
        stage_tile(cb_base, s_base, nt, lds_off(&cbT[buf ^ 1][tid]), lds_off(&sT[buf ^ 1][tid]));
        asm volatile("s_wait_asynccnt 0x2" ::: "memory");   // current tile landed
      } else {
        asm volatile("s_wait_asynccnt 0x0" ::: "memory");
      }
      float4 c  = cbT[buf][tid];
      float  sv = sT[buf][tid];
      int    i  = tile * BS + tid;
      if (prevOk && sv != NEGINF) {
        float ih = fminf(c.z, by2) - fmaxf(c.x, by1);
        float iw = fminf(c.w, bx2) - fmaxf(c.y, bx1);
        if (ih > 0.f && iw > 0.f) {
          float inter = ih * iw;
          float ai    = (c.z - c.x) * (c.w - c.y);
          // iou > 0.5  <=>  inter/(ai+ba-inter) > 0.5  <=>  3*inter > ai+ba
          if (3.f * inter > ai + ba) { sv = NEGINF; s[i] = NEGINF; }
        }
      }
      u64 k = enc_key(sv, i);
      best  = (k > best) ? k : best;
      tile  = nt;
      buf ^= 1;
    }

    // ---- Block reduce (wave32 butterflies + LDS), publish via atomicMax ----
    for (int o = 16; o > 0; o >>= 1) {
      unsigned lo = (unsigned)best, hi = (unsigned)(best >> 32);
      unsigned lo2 = __shfl_xor(lo, o, 32);
      unsigned hi2 = __shfl_xor(hi, o, 32);
      u64 other = ((u64)hi2 << 32) | (u64)lo2;
      best = (other > best) ? other : best;
    }
    if ((tid & 31) == 0) red[tid >> 5] = best;
    __syncthreads();
    if (tid == 0) {
      u64 b2 = red[0];
#pragma unroll
      for (int w = 1; w < NW; ++w) b2 = (red[w] > b2) ? red[w] : b2;
      __hip_atomic_fetch_max(&keys[t], b2, __ATOMIC_RELAXED, __HIP_MEMORY_SCOPE_AGENT);
    }

    gsync(bar, bar + 1);   // single grid sync per round

    // ---- Every block decodes the winner independently (L2 hit) ----
    if (tid == 0) {
      u64 k = __hip_atomic_load(&keys[t], __ATOMIC_RELAXED, __HIP_MEMORY_SCOPE_AGENT);
      float val; int idx;
      dec_key(k, val, idx);
      int ok = (val > NEGINF) ? 1 : 0;
      float4 cc = cb[idx];                // write-once data: no staleness hazard
      sel[0] = cc.x; sel[1] = cc.y; sel[2] = cc.z; sel[3] = cc.w;
      sel[4] = (cc.z - cc.x) * (cc.w - cc.y);
      selOk  = ok;
      if (blockIdx.x == 0) {
        float os = 0.f, ob0 = 0.f, ob1 = 0.f, ob2 = 0.f, ob3 = 0.f, oc = 0.f;
        if (ok && idx < n) {
          os = val;                       // == original score of the pick
          float4 rb = boxes[idx];         // raw (un-canonicalized) box
          ob0 = rb.x; ob1 = rb.y; ob2 = rb.z; ob3 = rb.w;
          oc  = (float)classes[idx];
        }
        out[t]                 = os;
        out[MAXB + 4 * t + 0]  = ob0;
        out[MAXB + 4 * t + 1]  = ob1;
        out[MAXB + 4 * t + 2]  = ob2;
        out[MAXB + 4 * t + 3]  = ob3;
        out[5 * MAXB + t]      = oc;
      }
    }
    __syncthreads();
    prevOk = (selOk != 0);
    by1 = sel[0]; bx1 = sel[1]; by2 = sel[2]; bx2 = sel[3]; ba = sel[4];
  }
}

extern "C" void kernel_launch(void* const* d_in, const int* in_sizes, int n_in,
                              void* d_out, int out_size, void* d_ws, size_t ws_size,
                              hipStream_t stream) {
  (void)n_in; (void)out_size;
  const float*  scores  = (const float*)d_in[0];
  const float4* boxes   = (const float4*)d_in[1];
  const int*    classes = (const int*)d_in[2];
  float*        out     = (float*)d_out;

  const int n     = in_sizes[0];
  const int tiles = (n + BS - 1) / BS;
  const int npad  = tiles * BS;

  // Workspace layout (all 16B-aligned since npad is a multiple of 256):
  //   float  s[npad]      : live (thresholded/suppressed) scores
  //   float4 cb[npad]     : canonicalized boxes [y1,x1,y2,x2]
  //   u64    keys[MAXB]   : per-round argmax keys
  //   u32    bar[2]       : grid-barrier {count, sense}
  char*     w    = (char*)d_ws;
  float*    s    = (float*)w;
  float4*   cb   = (float4*)(w + (size_t)npad * 4u);
  u64*      keys = (u64*)(w + (size_t)npad * 20u);
  unsigned* bar  = (unsigned*)(keys + MAXB);

  size_t need = (size_t)npad * 20u + (size_t)MAXB * 8u + 2u * sizeof(unsigned);
  if (ws_size < need) return;   // deterministic no-op if scratch is too small

  nms_init_kernel<<<1, BS, 0, stream>>>(keys, bar);
  nms_persistent_kernel<<<NB, BS, 0, stream>>>(scores, boxes, classes, out,
                                               s, cb, keys, bar, n, tiles);
}